// GCN_34368328302938
// MI455X (gfx1250) — compile-verified
//
#include <hip/hip_runtime.h>
#include <hip/hip_bf16.h>
#include <math.h>

// GCN forward on MI455X (gfx1250, wave32).
// Aggregate-then-transform (valid by linearity of segment_sum): both edge
// passes move only F=5 features/edge; dense transforms use
// V_WMMA_F32_16X16X4_F32 (16-node tiles per wave, K=5 -> 2 chained WMMAs).

typedef __attribute__((ext_vector_type(2))) float v2f;
typedef __attribute__((ext_vector_type(8))) float v8f;

#define F 5          // feature width on the edge path (F_IN == F_HID == 5)
#define F_OUT 8

// ---------------------------------------------------------------- degree ----
__global__ void k_init_deg(float* __restrict__ deg, int n) {
    int i = blockIdx.x * blockDim.x + threadIdx.x;
    if (i < n) deg[i] = 1.0f;                       // self-loop
}

__global__ void k_deg(const int* __restrict__ dst, float* __restrict__ deg, int e) {
    int i = blockIdx.x * blockDim.x + threadIdx.x;
    if (i < e) atomicAdd(&deg[dst[i]], 1.0f);
}

// dinv = rsqrt(deg) in place; seed agg1 with self-loop term x[i]*dinv[i]^2
__global__ void k_dinv_seed(const float* __restrict__ x, float* __restrict__ dinv_io,
                            float* __restrict__ agg1, int n) {
    int i = blockIdx.x * blockDim.x + threadIdx.x;
    if (i >= n) return;
    float di = __frsqrt_rn(dinv_io[i]);             // deg >= 1 always
    dinv_io[i] = di;
    float s = di * di;
    int ro = i * F;                                 // fits in 32 bits
    const float* xr = x + ro;
    float* ar = agg1 + ro;
#pragma unroll
    for (int j = 0; j < F; ++j) ar[j] = xr[j] * s;
}

// ------------------------------------------------------------ edge phase ----
// agg[dst] += feat[src] * dinv[src]*dinv[dst]   (5 features per edge)
__global__ void k_edge(const int* __restrict__ src, const int* __restrict__ dst,
                       const float* __restrict__ dinv, const float* __restrict__ feat,
                       float* __restrict__ agg, int e) {
    int i = blockIdx.x * blockDim.x + threadIdx.x;
    if (i >= e) return;
    int s = src[i], d = dst[i];
    float nm = dinv[s] * dinv[d];
    const float* fr = feat + s * F;                 // 32-bit offsets
    float* ar = agg + d * F;
#pragma unroll
    for (int j = 0; j < F; ++j) atomicAdd(&ar[j], fr[j] * nm);
}

// ------------------------------------------------- WMMA transform helper ----
// One wave owns a 16-node tile. A (16x4 f32): lanes 0-15 -> K=0,1 in v0,v1;
// lanes 16-31 -> K=2,3. B (4x16 f32): v0 = row(2*half), v1 = row(2*half+1),
// column = lane&15. C/D: lane holds rows M = r + 8*half, column N = lane&15.
template<bool FULL>
__device__ __forceinline__ v8f gcn_tile_gemm(const float* __restrict__ agg,
                                             const float* __restrict__ W,
                                             const float* __restrict__ bias,
                                             int tile, int n, int fout,
                                             int half, int col) {
    int node = tile + col;
    bool ok = FULL || (node < n);
    const float* ar = agg + (FULL ? node : (ok ? node : 0)) * F;
    int k0 = 2 * half;

    v2f a1, a2, b1v, b2v;
    if (FULL) {
        a1.x = ar[k0];
        a1.y = ar[k0 + 1];
        a2.x = (half == 0) ? ar[4] : 0.0f;
    } else {
        a1.x = ok ? ar[k0]     : 0.0f;
        a1.y = ok ? ar[k0 + 1] : 0.0f;
        a2.x = (ok && half == 0) ? ar[4] : 0.0f;
    }
    a2.y = 0.0f;                                    // padded K

    // clamped-index unconditional loads + select (no exec branches)
    bool cok = col < fout;
    int cw = cok ? col : 0;
    float w0 = W[k0 * fout + cw];
    float w1 = W[(k0 + 1) * fout + cw];
    float w4 = W[4 * fout + cw];
    float bb = bias[cw];
    b1v.x = cok ? w0 : 0.0f;
    b1v.y = cok ? w1 : 0.0f;
    b2v.x = (cok && half == 0) ? w4 : 0.0f;
    b2v.y = 0.0f;
    bb = cok ? bb : 0.0f;

    v8f c;
#pragma unroll
    for (int r = 0; r < 8; ++r) c[r] = bb;          // C[m][n] = bias[n]

    c = __builtin_amdgcn_wmma_f32_16x16x4_f32(false, a1, false, b1v, (short)0, c, false, false);
    c = __builtin_amdgcn_wmma_f32_16x16x4_f32(false, a2, false, b2v, (short)0, c, false, false);
    return c;
}

// h1 = relu(agg1 @ W1 + b1); also seed agg2 with layer-2 self-loop term
template<bool FULL>
__global__ void k_xform1(const float* __restrict__ agg1, const float* __restrict__ W1,
                         const float* __restrict__ b1, const float* __restrict__ dinv,
                         float* __restrict__ h1, float* __restrict__ agg2,
                         int tile0, int tile_end, int n) {
    int wave = threadIdx.x >> 5;
    int lane = threadIdx.x & 31;
    int half = lane >> 4;
    int col  = lane & 15;
    int tile = tile0 + (blockIdx.x * (blockDim.x >> 5) + wave) * 16;
    if (tile >= tile_end) return;                   // wave-uniform exit

    v8f c = gcn_tile_gemm<FULL>(agg1, W1, b1, tile, n, F, half, col);

    if (col < F) {
        int rbase = tile + 8 * half;                // first of this lane's 8 rows
        float* hp = h1   + rbase * F + col;
        float* ap = agg2 + rbase * F + col;
        const float* dp = dinv + rbase;
#pragma unroll
        for (int r = 0; r < 8; ++r) {
            if (FULL || (rbase + r < n)) {
                float v = fmaxf(c[r], 0.0f);        // relu
                float di = dp[r];
                hp[r * F] = v;                      // immediate-offset stores
                ap[r * F] = v * di * di;
            }
        }
    }
}

// out = log_softmax(agg2 @ W2 + b2) ; row reduction via LDS transpose tile
template<bool FULL>
__global__ void k_xform2(const float* __restrict__ agg2, const float* __restrict__ W2,
                         const float* __restrict__ b2, float* __restrict__ out,
                         int tile0, int tile_end, int n) {
    __shared__ float zt[8][16][F_OUT + 1];          // [wave][row][class]
    int wave = threadIdx.x >> 5;
    int lane = threadIdx.x & 31;
    int half = lane >> 4;
    int col  = lane & 15;
    int tile = tile0 + (blockIdx.x * (blockDim.x >> 5) + wave) * 16;

    if (tile < tile_end) {
        v8f c = gcn_tile_gemm<FULL>(agg2, W2, b2, tile, n, F_OUT, half, col);
        if (col < F_OUT) {
#pragma unroll
            for (int r = 0; r < 8; ++r)
                zt[wave][r + 8 * half][col] = c[r]; // transpose D into LDS
        }
    }
    __syncthreads();

    if (tile < tile_end && lane < 16) {
        int m = tile + lane;
        if (FULL || (m < n)) {
            float z[F_OUT];
            float mx = -INFINITY;
#pragma unroll
            for (int j = 0; j < F_OUT; ++j) { z[j] = zt[wave][lane][j]; mx = fmaxf(mx, z[j]); }
            float s = 0.0f;
#pragma unroll
            for (int j = 0; j < F_OUT; ++j) s += __expf(z[j] - mx);
            float lse = mx + __logf(s);
            float* orow = out + m * F_OUT;
#pragma unroll
            for (int j = 0; j < F_OUT; ++j) orow[j] = z[j] - lse;
        }
    }
}

// ------------------------------------------------------------------ launch --
extern "C" void kernel_launch(void* const* d_in, const int* in_sizes, int n_in,
                              void* d_out, int out_size, void* d_ws, size_t ws_size,
                              hipStream_t stream) {
    const float* x  = (const float*)d_in[0];
    const int*   ei = (const int*)  d_in[1];
    const float* W1 = (const float*)d_in[2];
    const float* b1 = (const float*)d_in[3];
    const float* W2 = (const float*)d_in[4];
    const float* b2 = (const float*)d_in[5];
    float* out = (float*)d_out;

    const int n = in_sizes[0] / F;      // 500,000
    const int e = in_sizes[1] / 2;      // 16,000,000
    const int* src = ei;
    const int* dst = ei + e;

    float* ws   = (float*)d_ws;         // needs 16*n floats (~32 MB)
    float* dinv = ws;                   // n      (deg, then rsqrt in place)
    float* agg1 = ws + (size_t)n;       // 5n
    float* h1   = ws + (size_t)6 * n;   // 5n
    float* agg2 = ws + (size_t)11 * n;  // 5n

    const int BT = 256;
    const int NPB = (BT / 32) * 16;     // nodes per block in xform (128)
    int gn = (n + BT - 1) / BT;
    int ge = (e + BT - 1) / BT;

    int n_full   = (n / 16) * 16;       // full-tile region (== n when n%16==0)
    int gx_full  = (n_full + NPB - 1) / NPB;
    bool tail    = (n_full < n);

    k_init_deg <<<gn, BT, 0, stream>>>(dinv, n);
    k_deg      <<<ge, BT, 0, stream>>>(dst, dinv, e);
    k_dinv_seed<<<gn, BT, 0, stream>>>(x, dinv, agg1, n);

    k_edge     <<<ge, BT, 0, stream>>>(src, dst, dinv, x, agg1, e);
    if (gx_full)
        k_xform1<true> <<<gx_full, BT, 0, stream>>>(agg1, W1, b1, dinv, h1, agg2, 0, n_full, n);
    if (tail)
        k_xform1<false><<<1, BT, 0, stream>>>(agg1, W1, b1, dinv, h1, agg2, n_full, n, n);

    k_edge     <<<ge, BT, 0, stream>>>(src, dst, dinv, h1, agg2, e);
    if (gx_full)
        k_xform2<true> <<<gx_full, BT, 0, stream>>>(agg2, W2, b2, out, 0, n_full, n);
    if (tail)
        k_xform2<false><<<1, BT, 0, stream>>>(agg2, W2, b2, out, n_full, n, n);
}